// SphericalExpansion_33535104647552
// MI455X (gfx1250) — compile-verified
//
#include <hip/hip_runtime.h>

typedef float v2f __attribute__((ext_vector_type(2)));
typedef float v8f __attribute__((ext_vector_type(8)));

#define WAVES_PER_BLOCK 4
#define GS_STRIDE 17   // 32x16 tile padded -> conflict-free column reads
#define RS_STRIDE 33   // 32x32 tile padded

__global__ __launch_bounds__(128) void sph_expand_wmma(
    const float* __restrict__ dist,
    const float* __restrict__ dirs,
    const float* __restrict__ W,        // [4][8][16] l-major, k fastest
    const float* __restrict__ centers,  // [16]
    const int*   __restrict__ zarr,     // [N_ATOMS]
    const int*   __restrict__ idx_i,
    const int*   __restrict__ idx_j,
    float*       __restrict__ out,      // [N_ATOMS*4*8*16]
    int n_edges)
{
    __shared__ float g_s[WAVES_PER_BLOCK][32 * GS_STRIDE];
    __shared__ float Y_s[WAVES_PER_BLOCK][32 * GS_STRIDE];
    __shared__ float R_s[WAVES_PER_BLOCK][32 * RS_STRIDE];
    __shared__ int   b_s[WAVES_PER_BLOCK][32];

    const int lane = threadIdx.x & 31;
    const int wv   = threadIdx.x >> 5;
    const int hi   = lane >> 4;      // half-wave: 0 or 1
    const int lo   = lane & 15;

    const int batch = blockIdx.x * WAVES_PER_BLOCK + wv;
    if (batch * 32 + 32 > n_edges) return;   // whole-wave guard (exact fit here)
    const int e = batch * 32 + lane;

    // ---------------- B fragments (W matrix) into registers ----------------
    // B is K(4) x N(16) per (kc, nt). Assumed mirror of A layout:
    //   VGPR0: lanes0-15 K=+0, lanes16-31 K=+2 ; VGPR1: K=+1 / K=+3
    // column col = l*8+n  ->  W flat index col*16 + k
    v2f Bfrag[2][4];
#pragma unroll
    for (int nt = 0; nt < 2; ++nt) {
        const int col = nt * 16 + lo;
#pragma unroll
        for (int kc = 0; kc < 4; ++kc) {
            const int kb = kc * 4 + hi * 2;
            v2f b;
            b.x = W[col * 16 + kb];
            b.y = W[col * 16 + kb + 1];
            Bfrag[nt][kc] = b;
        }
    }

    // ---------------- per-edge radial basis (lane = edge) ----------------
    const float r = dist[e];
    const float t  = (r - 4.5f) * 2.0f;                      // (r-(RC-SW))/SW
    const float tc = fminf(fmaxf(t, 0.0f), 1.0f);
    const float ramp = 0.5f * (cosf(3.14159265358979323846f * tc) + 1.0f);
    const float fc = (r < 4.5f) ? 1.0f : ((r < 5.0f) ? ramp : 0.0f);

    float* gr = &g_s[wv][lane * GS_STRIDE];
#pragma unroll
    for (int k = 0; k < 16; ++k) {
        const float d = r - centers[k];
        gr[k] = fc * expf(-2.0f * d * d);     // fold cutoff into g
    }

    // ---------------- A fragments: ISA 16x4 f32 layout ----------------
    // lanes0-15: M=lane, VGPR0=K0,VGPR1=K1 ; lanes16-31: M=lane-16, K2/K3
    v2f Afrag[2][4];
#pragma unroll
    for (int mt = 0; mt < 2; ++mt) {
        const float* gp = &g_s[wv][(mt * 16 + lo) * GS_STRIDE];
#pragma unroll
        for (int kc = 0; kc < 4; ++kc) {
            const int ka = kc * 4 + hi * 2;
            v2f a;
            a.x = gp[ka];
            a.y = gp[ka + 1];
            Afrag[mt][kc] = a;
        }
    }

    // ---------------- 16x V_WMMA_F32_16X16X4_F32 ----------------
    // R[32 edges][32 cols] = g(32x16) * W^T(16x32), K accumulated over 4 chunks
#pragma unroll
    for (int mt = 0; mt < 2; ++mt) {
#pragma unroll
        for (int nt = 0; nt < 2; ++nt) {
            v8f acc = {0.f, 0.f, 0.f, 0.f, 0.f, 0.f, 0.f, 0.f};
#pragma unroll
            for (int kc = 0; kc < 4; ++kc) {
                acc = __builtin_amdgcn_wmma_f32_16x16x4_f32(
                    false, Afrag[mt][kc], false, Bfrag[nt][kc],
                    (short)0, acc, false, false);
            }
            // D layout: lane = N (lo), VGPR j -> M = j + 8*hi
            const int erow = mt * 16 + hi * 8;
            const int col  = nt * 16 + lo;
#pragma unroll
            for (int j = 0; j < 8; ++j)
                R_s[wv][(erow + j) * RS_STRIDE + col] = acc[j];
        }
    }

    // ---------------- Ylm + scatter base (lane = edge) ----------------
    {
        const float x  = dirs[e * 3 + 0];
        const float y  = dirs[e * 3 + 1];
        const float zc = dirs[e * 3 + 2];
        const float x2 = x * x, y2 = y * y, z2 = zc * zc;
        float* Yr = &Y_s[wv][lane * GS_STRIDE];
        Yr[0]  = 0.28209479177387814f;
        Yr[1]  = 0.4886025119029199f * y;
        Yr[2]  = 0.4886025119029199f * zc;
        Yr[3]  = 0.4886025119029199f * x;
        Yr[4]  = 1.0925484305920792f * x * y;
        Yr[5]  = 1.0925484305920792f * y * zc;
        Yr[6]  = 0.31539156525252005f * (3.0f * z2 - 1.0f);
        Yr[7]  = 1.0925484305920792f * x * zc;
        Yr[8]  = 0.5462742152960396f * (x2 - y2);
        Yr[9]  = 0.5900435899266435f * y * (3.0f * x2 - y2);
        Yr[10] = 2.890611442640554f * x * y * zc;
        Yr[11] = 0.4570457994644658f * y * (5.0f * z2 - 1.0f);
        Yr[12] = 0.3731763325901154f * zc * (5.0f * z2 - 3.0f);
        Yr[13] = 0.4570457994644658f * x * (5.0f * z2 - 1.0f);
        Yr[14] = 1.445305721320277f * zc * (x2 - y2);
        Yr[15] = 0.5900435899266435f * x * (x2 - 3.0f * y2);

        b_s[wv][lane] = (idx_i[e] * 4 + zarr[idx_j[e]]) * 128;
    }

    // ---------------- coalesced scatter-add (lane = channel) ----------------
    // channel c = n*16 + lm ; lm = lane&15 (constant per lane), n = ch*2 + hi
    const int lm = lo;
    const int l  = (lm >= 9) ? 3 : (lm >= 4) ? 2 : (lm >= 1) ? 1 : 0;

    for (int le = 0; le < 32; ++le) {
        const float yv   = Y_s[wv][le * GS_STRIDE + lm];
        float* op        = out + b_s[wv][le];
        const float* rp  = &R_s[wv][le * RS_STRIDE + l * 8];
#pragma unroll
        for (int ch = 0; ch < 4; ++ch) {
            const float v = rp[ch * 2 + hi] * yv;
            atomicAdd(op + ch * 32 + lane, v);   // 32 consecutive floats / instr
        }
    }
}

extern "C" void kernel_launch(void* const* d_in, const int* in_sizes, int n_in,
                              void* d_out, int out_size, void* d_ws, size_t ws_size,
                              hipStream_t stream) {
    const float* dist    = (const float*)d_in[0];
    const float* dirs    = (const float*)d_in[1];
    const float* W       = (const float*)d_in[2];
    const float* centers = (const float*)d_in[3];
    const int*   zarr    = (const int*)d_in[4];
    const int*   idx_i   = (const int*)d_in[5];
    const int*   idx_j   = (const int*)d_in[6];
    float* out = (float*)d_out;

    const int n_edges = in_sizes[0];

    // atomics accumulate onto zeros; re-done every call -> graph-replay safe
    hipMemsetAsync(out, 0, (size_t)out_size * sizeof(float), stream);

    const int batches = (n_edges + 31) / 32;                          // 25000
    const int blocks  = (batches + WAVES_PER_BLOCK - 1) / WAVES_PER_BLOCK; // 6250
    sph_expand_wmma<<<blocks, 128, 0, stream>>>(
        dist, dirs, W, centers, zarr, idx_i, idx_j, out, n_edges);
}